// GCN_60163901882497
// MI455X (gfx1250) — compile-verified
//
#include <hip/hip_runtime.h>
#include <hip/hip_bf16.h>

typedef float v2f __attribute__((ext_vector_type(2)));
typedef float v8f __attribute__((ext_vector_type(8)));

#define N_NODES    100000
#define N_EDGES    1000000
#define N_FEAT     136
#define HIDDEN     64
#define N_CLASSES  2
#define NUM_GRAPHS 512

static inline int ceilDiv(long long a, long long b) { return (int)((a + b - 1) / b); }

// ---------------- small utility kernels ----------------

__global__ void k_fill(float* __restrict__ p, float v, int n) {
    int i = blockIdx.x * blockDim.x + threadIdx.x;
    if (i < n) p[i] = v;
}

// deg[dst[e]] += 1  (deg pre-initialized to 1.0 for the self-loop)
__global__ void k_deg_count(const int* __restrict__ dst, float* __restrict__ deg, int e) {
    int i = blockIdx.x * blockDim.x + threadIdx.x;
    if (i < e) atomicAdd(&deg[dst[i]], 1.0f);
}

// dinv = rsqrt(deg) in place (deg >= 1 always due to self-loops)
__global__ void k_rsqrt(float* __restrict__ p, int n) {
    int i = blockIdx.x * blockDim.x + threadIdx.x;
    if (i < n) p[i] = rsqrtf(p[i]);
}

// Seed the aggregation buffer with bias + the self-loop contribution:
//   out[n][f] = b[f] + h[n][f] * dinv[n]^2
// Self-loop destinations are unique per node -> plain (non-atomic) float4 writes.
__global__ void k_seed(const float* __restrict__ h, const float* __restrict__ dinv,
                       const float* __restrict__ b, float* __restrict__ out) {
    int i = blockIdx.x * blockDim.x + threadIdx.x;     // over N_NODES * 16 quads
    if (i >= N_NODES * 16) return;
    const int n = i >> 4;
    const int q = i & 15;
    float w = dinv[n];
    w = w * w;
    const float4 hv = reinterpret_cast<const float4*>(h + (size_t)n * HIDDEN)[q];
    const float4 bv = reinterpret_cast<const float4*>(b)[q];
    float4 o;
    o.x = bv.x + hv.x * w;
    o.y = bv.y + hv.y * w;
    o.z = bv.z + hv.z * w;
    o.w = bv.w + hv.w * w;
    reinterpret_cast<float4*>(out + (size_t)n * HIDDEN)[q] = o;
}

// ---------------- WMMA fp32 GEMM:  D[nRows,64] = (relu?)A[nRows,K] @ W[K,64] ----------------
// One wave computes a 16x64 output strip using V_WMMA_F32_16X16X4_F32.
// A layout (16x4 f32): lane l holds row (l&15), K = (l>>4)*2 + {0,1} in 2 VGPRs.
// B layout (4x16 f32): lane l holds col (l&15), K = (l>>4)*2 + {0,1} in 2 VGPRs.
// C/D (16x16 f32): VGPR v holds (M = v + 8*(l>>4), N = l&15).
template <int K, bool RELU_A>
__global__ void __launch_bounds__(256)
k_gemm_wmma(const float* __restrict__ A, const float* __restrict__ W,
            float* __restrict__ D, int nRows) {
    __shared__ float sW[K * HIDDEN];
    for (int i = threadIdx.x; i < K * HIDDEN; i += 256) sW[i] = W[i];
    __syncthreads();

    const int wave = threadIdx.x >> 5;           // 8 waves / block (wave32)
    const int lane = threadIdx.x & 31;
    const int tile = blockIdx.x * 8 + wave;      // 16-row tile index
    if (tile * 16 >= nRows) return;              // nRows % 16 == 0, whole-wave exit

    const int r  = lane & 15;
    const int hi = lane >> 4;

    const float* Arow = A + (size_t)(tile * 16 + r) * K;

    v8f acc[4] = {v8f{}, v8f{}, v8f{}, v8f{}};

    for (int kk = 0; kk < K; kk += 4) {
        const int k0 = kk + hi * 2;
        float a0 = Arow[k0];
        float a1 = Arow[k0 + 1];
        if (RELU_A) { a0 = fmaxf(a0, 0.0f); a1 = fmaxf(a1, 0.0f); }
        v2f a; a.x = a0; a.y = a1;
#pragma unroll
        for (int cb = 0; cb < 4; ++cb) {
            v2f b;
            b.x = sW[k0 * HIDDEN + cb * 16 + r];
            b.y = sW[(k0 + 1) * HIDDEN + cb * 16 + r];
            acc[cb] = __builtin_amdgcn_wmma_f32_16x16x4_f32(
                false, a, false, b, (short)0, acc[cb], false, false);
        }
    }

#pragma unroll
    for (int cb = 0; cb < 4; ++cb) {
#pragma unroll
        for (int v = 0; v < 8; ++v) {
            const int row = tile * 16 + v + hi * 8;
            D[(size_t)row * HIDDEN + cb * 16 + r] = acc[cb][v];
        }
    }
}

// ---------------- edge propagation: out[dst] += h[src] * dinv[src]*dinv[dst] ----------------
// Real edges only (self-loops handled in k_seed).
// One thread = one (edge, feature-quad); float4 gather + 4 global f32 atomics.
__global__ void k_scatter(const float* __restrict__ h, const float* __restrict__ dinv,
                          const int* __restrict__ src, const int* __restrict__ dst,
                          float* __restrict__ out, int nE) {
    long long idx = (long long)blockIdx.x * blockDim.x + threadIdx.x;
    if (idx >= (long long)nE * 16) return;
    const int e = (int)(idx >> 4);
    const int q = (int)(idx & 15);
    const int s = src[e];
    const int d = dst[e];
    const float norm = dinv[s] * dinv[d];
    const float4 v = reinterpret_cast<const float4*>(h + (size_t)s * HIDDEN)[q];
    float* o = out + (size_t)d * HIDDEN + q * 4;
    atomicAdd(o + 0, v.x * norm);
    atomicAdd(o + 1, v.y * norm);
    atomicAdd(o + 2, v.z * norm);
    atomicAdd(o + 3, v.w * norm);
}

// ---------------- pooling: segment sums of relu(h) + counts ----------------
__global__ void k_pool(const float* __restrict__ h, const int* __restrict__ batch,
                       float* __restrict__ pooled, float* __restrict__ counts) {
    int i = blockIdx.x * blockDim.x + threadIdx.x;
    if (i >= N_NODES * HIDDEN) return;
    const int node = i >> 6;
    const int f    = i & (HIDDEN - 1);
    const int g    = batch[node];
    atomicAdd(&pooled[g * HIDDEN + f], fmaxf(h[i], 0.0f));
    if (f == 0) atomicAdd(&counts[g], 1.0f);
}

// ---------------- head: out[g,c] = (pooled[g]/max(cnt,1)) . Wout[:,c] + bout[c] ----------------
__global__ void k_head(const float* __restrict__ pooled, const float* __restrict__ counts,
                       const float* __restrict__ Wout, const float* __restrict__ bout,
                       float* __restrict__ out) {
    int i = blockIdx.x * blockDim.x + threadIdx.x;
    if (i >= NUM_GRAPHS * N_CLASSES) return;
    const int g = i / N_CLASSES;
    const int c = i % N_CLASSES;
    const float inv = 1.0f / fmaxf(counts[g], 1.0f);
    float s = bout[c];
    const float* p = pooled + g * HIDDEN;
#pragma unroll 8
    for (int f = 0; f < HIDDEN; ++f) s += (p[f] * inv) * Wout[f * N_CLASSES + c];
    out[i] = s;
}

// ---------------- launch ----------------
extern "C" void kernel_launch(void* const* d_in, const int* in_sizes, int n_in,
                              void* d_out, int out_size, void* d_ws, size_t ws_size,
                              hipStream_t stream) {
    const float* x    = (const float*)d_in[0];
    const int*   ei   = (const int*)d_in[1];           // [2, E] flat
    const int*   srcI = ei;
    const int*   dstI = ei + N_EDGES;
    const int*   batch= (const int*)d_in[2];
    const float* W1   = (const float*)d_in[3];
    const float* b1   = (const float*)d_in[4];
    const float* W2   = (const float*)d_in[5];
    const float* b2   = (const float*)d_in[6];
    const float* Wout = (const float*)d_in[7];
    const float* bout = (const float*)d_in[8];
    float* out = (float*)d_out;

    // workspace layout (floats); offsets keep 16B alignment for float4 access
    float* ws     = (float*)d_ws;
    float* dinv   = ws;                                  // N
    float* bufA   = ws   + N_NODES;                      // N*64 (100000 % 16 == 0)
    float* bufB   = bufA + (size_t)N_NODES * HIDDEN;     // N*64
    float* pooled = bufB + (size_t)N_NODES * HIDDEN;     // 512*64
    float* counts = pooled + NUM_GRAPHS * HIDDEN;        // 512

    const int BS = 256;

    // degrees -> dinv
    k_fill<<<ceilDiv(N_NODES, BS), BS, 0, stream>>>(dinv, 1.0f, N_NODES);
    k_deg_count<<<ceilDiv(N_EDGES, BS), BS, 0, stream>>>(dstI, dinv, N_EDGES);
    k_rsqrt<<<ceilDiv(N_NODES, BS), BS, 0, stream>>>(dinv, N_NODES);

    const int gemmBlocks     = ceilDiv(N_NODES / 16, 8);          // 8 waves/block, 16 rows/wave
    const int seedBlocks     = ceilDiv((long long)N_NODES * 16, BS);
    const int scatterBlocks  = ceilDiv((long long)N_EDGES * 16, BS);

    // layer 1: h1 = X @ W1 ; agg1 = b1 + selfloop + scatter(norm * h1[src])
    k_gemm_wmma<N_FEAT, false><<<gemmBlocks, BS, 0, stream>>>(x, W1, bufA, N_NODES);
    k_seed<<<seedBlocks, BS, 0, stream>>>(bufA, dinv, b1, bufB);
    k_scatter<<<scatterBlocks, BS, 0, stream>>>(bufA, dinv, srcI, dstI, bufB, N_EDGES);

    // layer 2: h2 = relu(agg1) @ W2 ; agg2 = b2 + selfloop + scatter(norm * h2[src])
    k_gemm_wmma<HIDDEN, true><<<gemmBlocks, BS, 0, stream>>>(bufB, W2, bufA, N_NODES);
    k_seed<<<seedBlocks, BS, 0, stream>>>(bufA, dinv, b2, bufB);
    k_scatter<<<scatterBlocks, BS, 0, stream>>>(bufA, dinv, srcI, dstI, bufB, N_EDGES);

    // mean-pool relu(agg2) over graphs, then linear head
    k_fill<<<ceilDiv(NUM_GRAPHS * HIDDEN + NUM_GRAPHS, BS), BS, 0, stream>>>(
        pooled, 0.0f, NUM_GRAPHS * HIDDEN + NUM_GRAPHS);
    k_pool<<<ceilDiv((long long)N_NODES * HIDDEN, BS), BS, 0, stream>>>(bufB, batch, pooled, counts);
    k_head<<<ceilDiv(NUM_GRAPHS * N_CLASSES, BS), BS, 0, stream>>>(pooled, counts, Wout, bout, out);
}